// ExportableCostVolume_30614526886283
// MI455X (gfx1250) — compile-verified
//
#include <hip/hip_runtime.h>

typedef __attribute__((ext_vector_type(2))) float        v2f;
typedef __attribute__((ext_vector_type(8))) float        v8f;
typedef __attribute__((ext_vector_type(4))) unsigned int v4u;
typedef __attribute__((ext_vector_type(8))) int          v8i;
typedef __attribute__((ext_vector_type(4))) int          v4i;

#define Bz 4
#define Cc 128
#define Hh 256
#define Ww 512
#define Dd 128
#define NTHREADS 288   // 9 waves of 32

#if defined(__has_builtin)
#if __has_builtin(__builtin_amdgcn_tensor_load_to_lds) && \
    __has_builtin(__builtin_amdgcn_s_wait_tensorcnt)
#define USE_TDM 1
#endif
#endif
#ifndef USE_TDM
#define USE_TDM 0
#endif

// One workgroup = one (b, h, w-tile of 16). Wave t (0..8) computes the 16x16
// Gram tile G[wtile i, w'tile j=i-t] with V_WMMA_F32_16X16X4_F32, K = C = 128.
__global__ __launch_bounds__(NTHREADS)
void cost_volume_wmma_f32(const float* __restrict__ left,
                          const float* __restrict__ right,
                          float* __restrict__ out) {
    __shared__ float lds_left [Cc][16];    // L^T tile: [c][w_rel]       8 KB
    __shared__ float lds_right[Cc][144];   // R slab:   [c][w'_rel]     72 KB
    __shared__ float lds_out  [Dd][16];    // out tile: [d][w_rel]       8 KB

    const int i    = blockIdx.x;           // w-tile index (0..31)
    const int h    = blockIdx.y;
    const int b    = blockIdx.z;
    const int tid  = threadIdx.x;
    const int wave = tid >> 5;             // 0..8 = diagonal t
    const int lane = tid & 31;
    const int ln   = lane & 15;

    const size_t planeHW = (size_t)Hh * Ww;   // 131072 (c-stride and d-stride)
    const float* lbase = left  + ((size_t)b * Cc) * planeHW + (size_t)h * Ww + i * 16;
    const float* rbase = right + ((size_t)b * Cc) * planeHW + (size_t)h * Ww;
    const int wslab0 = i * 16 - 128;       // w' of slab column 0 (16-aligned, may be <0)

#if USE_TDM
    // ---- left tile via Tensor Data Mover: 2D tile 16 (w) x 128 (c rows), ----
    // ---- rows packed 16 floats each == lds_left[c][16] layout exactly.    ----
    if (wave == 0) {
        unsigned           lds_byte = (unsigned)(size_t)(void*)&lds_left[0][0];
        unsigned long long ga       = (unsigned long long)(size_t)lbase;
        v4u g0;
        g0.x = 1u;                                      // count=1 (valid user D#)
        g0.y = lds_byte;                                // lds_addr
        g0.z = (unsigned)(ga & 0xFFFFFFFFull);          // global_addr[31:0]
        g0.w = (unsigned)((ga >> 32) & 0x01FFFFFFull)   // global_addr[56:32]
             | (2u << 30);                              // type=2 ("image")
        const unsigned           td0 = 0x7FFFFFFFu;     // huge tensor dims: no OOB clip
        const unsigned           td1 = 0x7FFFFFFFu;
        const unsigned           tl0 = 16u;             // tile_dim0: 16 floats along w
        const unsigned           tl1 = (unsigned)Cc;    // tile_dim1: 128 c-rows
        const unsigned long long st0 = (unsigned long long)Hh * Ww; // row stride (elems)
        v8i g1;
        g1[0] = (int)(2u << 16);                              // data_size=2 -> 4 bytes
        g1[1] = (int)((td0 & 0xFFFFu) << 16);                 // abar_addr=0 | td0[15:0]
        g1[2] = (int)((td0 >> 16) | ((td1 & 0xFFFFu) << 16)); // td0[31:16] | td1[15:0]
        g1[3] = (int)((td1 >> 16) | (tl0 << 16));             // td1[31:16] | tile_dim0
        g1[4] = (int)tl1;                                     // tile_dim1 | tile_dim2=0
        g1[5] = (int)(st0 & 0xFFFFFFFFull);                   // stride0[31:0]
        g1[6] = (int)((st0 >> 32) & 0xFFFFull);               // stride0[47:32] | s1[15:0]=0
        g1[7] = 0;
        v4i gz = {0, 0, 0, 0};
#if defined(__clang_major__) && (__clang_major__ >= 23)
        v8i gz8 = {0, 0, 0, 0, 0, 0, 0, 0};
        __builtin_amdgcn_tensor_load_to_lds(g0, g1, gz, gz, gz8, 0);
#else
        __builtin_amdgcn_tensor_load_to_lds(g0, g1, gz, gz, 0);
#endif
    }
#else
    // ---- stage L^T tile [128 c x 16 w] manually ----
    for (int idx = tid; idx < Cc * 16; idx += NTHREADS) {
        int c = idx >> 4, wr = idx & 15;
        lds_left[c][wr] = lbase[(size_t)c * planeHW + wr];
    }
#endif

    // ---- stage R slab [128 c x 144 w'] as float4 (rows 576B -> aligned) ----
    for (int q = tid; q < Cc * 36; q += NTHREADS) {
        int c  = q / 36;
        int qq = q - c * 36;
        int wp = wslab0 + (qq << 2);       // whole quad in/out of range (16-aligned base)
        float4 v = make_float4(0.f, 0.f, 0.f, 0.f);
        if (wp >= 0)
            v = *(const float4*)(rbase + (size_t)c * planeHW + wp);
        *(float4*)(&lds_right[c][qq << 2]) = v;
    }
    // ---- zero (d,w) staging: covers the w<d invalid region automatically ----
    for (int idx = tid; idx < Dd * 16; idx += NTHREADS)
        ((float*)lds_out)[idx] = 0.0f;

#if USE_TDM
    if (wave == 0)
        __builtin_amdgcn_s_wait_tensorcnt(0);   // TDM done before barrier signal
#endif
    __syncthreads();

    const int t = wave;
    const int j = i - t;                   // right w'-tile index
    if (j >= 0) {                          // uniform per wave -> EXEC stays all-ones
        const int wrel0 = (8 - t) << 4;    // this wave's w'-tile offset in the slab
        const int khi   = (lane >> 4) << 1; // K base: 0 (lanes 0-15) / 2 (lanes 16-31)
        v8f acc = {0.f, 0.f, 0.f, 0.f, 0.f, 0.f, 0.f, 0.f};
#pragma unroll
        for (int k0 = 0; k0 < Cc; k0 += 4) {
            v2f a, bmat;
            // A (16x4, f32): VGPR0 = K {0|2}, VGPR1 = K {1|3}, M = lane&15
            a.x    = lds_left [k0 + khi    ][ln];
            a.y    = lds_left [k0 + khi + 1][ln];
            // B (4x16, f32): same half-wave K striping, N = lane&15
            bmat.x = lds_right[k0 + khi    ][wrel0 + ln];
            bmat.y = lds_right[k0 + khi + 1][wrel0 + ln];
            acc = __builtin_amdgcn_wmma_f32_16x16x4_f32(
                      false, a, false, bmat, (short)0, acc, false, false);
        }
        // D[m][n] = G[16i+m, 16j+n]  ->  d = 16t + m - n, w_rel = m, scale 1/C
        const float s     = 1.0f / 128.0f;
        const int   mbase = (lane >> 4) << 3;   // 0 or 8
        if (t >= 1 && t <= 7) {
            // interior diagonals: d = 16t + m - ln always in [1,127]
#pragma unroll
            for (int r = 0; r < 8; ++r) {
                int m = mbase + r;
                lds_out[(t << 4) + m - ln][m] = acc[r] * s;
            }
        } else {
#pragma unroll
            for (int r = 0; r < 8; ++r) {
                int m = mbase + r;
                int d = (t << 4) + m - ln;
                if (d >= 0 && d < Dd)
                    lds_out[d][m] = acc[r] * s;  // unique (d,m) across all waves
            }
        }
    }

    __syncthreads();

    // ---- coalesced non-temporal store: 16 consecutive w per 16-lane group ----
    float* obase = out + (((size_t)b * Dd) * Hh + h) * Ww + i * 16;
    for (int idx = tid; idx < Dd * 16; idx += NTHREADS) {
        int d = idx >> 4, wr = idx & 15;
        __builtin_nontemporal_store(((const float*)lds_out)[idx],
                                    obase + (size_t)d * planeHW + wr);
    }
}

extern "C" void kernel_launch(void* const* d_in, const int* in_sizes, int n_in,
                              void* d_out, int out_size, void* d_ws, size_t ws_size,
                              hipStream_t stream) {
    (void)in_sizes; (void)n_in; (void)out_size; (void)d_ws; (void)ws_size;
    const float* left  = (const float*)d_in[0];
    const float* right = (const float*)d_in[1];
    float*       out   = (float*)d_out;

    dim3 grid(Ww / 16, Hh, Bz);   // (32, 256, 4): w-tiles fastest -> L2 reuse of right slabs
    dim3 block(NTHREADS);
    cost_volume_wmma_f32<<<grid, block, 0, stream>>>(left, right, out);
}